// GNNPolicy_72688026517645
// MI455X (gfx1250) — compile-verified
//
#include <hip/hip_runtime.h>
#include <hip/hip_bf16.h>
#include <float.h>

#define NN 50000     // nodes (multiple of 16 -> 3125 row tiles, no tail)
#define NE 640000    // edges
#define OBS 64
#define HID 128
#define ACT 8

typedef __attribute__((ext_vector_type(2))) float v2f;
typedef __attribute__((ext_vector_type(8))) float v8f;

// ---------------------------------------------------------------------------
// GEMM: OUT[NN x 128] = X[NN x K] @ W[K x 128] + b   (optional fused ReLU)
// One wave per 16x16 output tile using V_WMMA_F32_16X16X4_F32.
// Block = 256 threads = 8 waves = 8 column tiles of one 16-row stripe.
// Fragment layouts per CDNA5 ISA 7.12.2:
//   A 16x4 f32 : lane m=lane&15 is row M; VGPR0/1 = K+kh, K+kh+1, kh=(lane>>4)*2
//   B 4x16 f32 : lane n=lane&15 is col N; VGPR0/1 = K+kh, K+kh+1
//   C/D 16x16  : VGPR j holds M = j + (lane>=16 ? 8 : 0), N = lane&15
// ---------------------------------------------------------------------------
template <int K, bool RELU>
__global__ __launch_bounds__(256) void gemm_wmma_f32(
    const float* __restrict__ X, const float* __restrict__ W,
    const float* __restrict__ b, float* __restrict__ OUT) {
  const int lane = threadIdx.x & 31;
  const int wv   = threadIdx.x >> 5;      // column tile 0..7
  const int row0 = blockIdx.x * 16;
  const int col0 = wv * 16;
  const int m    = lane & 15;             // A-row / B-col / D-col
  const int kh   = (lane >> 4) * 2;       // K sub-offset for upper half-wave

  v8f acc = {0.f, 0.f, 0.f, 0.f, 0.f, 0.f, 0.f, 0.f};
  const float* xrow = X + (size_t)(row0 + m) * K;
  for (int k = 0; k < K; k += 4) {
    v2f a, bt;
    a.x  = xrow[k + kh];
    a.y  = xrow[k + kh + 1];
    bt.x = W[(size_t)(k + kh) * HID + col0 + m];
    bt.y = W[(size_t)(k + kh + 1) * HID + col0 + m];
    acc = __builtin_amdgcn_wmma_f32_16x16x4_f32(
        /*neg_a=*/false, a, /*neg_b=*/false, bt,
        /*c_mod=*/(short)0, acc, /*reuse_a=*/false, /*reuse_b=*/false);
  }
  const int   mb   = (lane >> 4) * 8;
  const float bias = b[col0 + m];
#pragma unroll
  for (int j = 0; j < 8; ++j) {
    float v = acc[j] + bias;
    if (RELU) v = fmaxf(v, 0.f);
    OUT[(size_t)(row0 + mb + j) * HID + col0 + m] = v;
  }
}

// ---------------------------------------------------------------------------
// Segment-softmax scratch init: agg=0 over NN*HID, smax=-FLT_MAX, denom=0.
// Grid covers exactly NN*HID.
// ---------------------------------------------------------------------------
__global__ void init_seg(float* __restrict__ smax, float* __restrict__ denom,
                         float* __restrict__ agg) {
  const int i = blockIdx.x * blockDim.x + threadIdx.x;
  agg[i] = 0.f;
  if (i < NN) { smax[i] = -FLT_MAX; denom[i] = 0.f; }
}

// Monotonic float atomic max (init value is -FLT_MAX).
__device__ inline void atomicMaxF(float* addr, float v) {
  if (v >= 0.f) atomicMax((int*)addr, __float_as_int(v));
  else          atomicMin((unsigned int*)addr, __float_as_uint(v));
}

// ---------------------------------------------------------------------------
// Pass A: per-edge attention logit  s = att . leaky_relu(xl[src]+xr[dst]+ea*We)
// One wave per edge; lane handles 4 channels (float4); wave-reduce via shfl.
// ---------------------------------------------------------------------------
__global__ __launch_bounds__(256) void edge_logits(
    const int* __restrict__ src, const int* __restrict__ dst,
    const float* __restrict__ ea, const float* __restrict__ xl,
    const float* __restrict__ xr, const float* __restrict__ We,
    const float* __restrict__ att, float* __restrict__ sbuf,
    float* __restrict__ smax) {
  const int e    = blockIdx.x * 8 + (threadIdx.x >> 5);
  const int lane = threadIdx.x & 31;
  const int s = src[e], d = dst[e];
  const float a = ea[e];
  const float4 xlv = *(const float4*)(xl + (size_t)s * HID + lane * 4);
  const float4 xrv = *(const float4*)(xr + (size_t)d * HID + lane * 4);
  const float4 wev = *(const float4*)(We + lane * 4);
  const float4 atv = *(const float4*)(att + lane * 4);
  float p = 0.f, mm;
  mm = xlv.x + xrv.x + a * wev.x; p += atv.x * (mm > 0.f ? mm : 0.2f * mm);
  mm = xlv.y + xrv.y + a * wev.y; p += atv.y * (mm > 0.f ? mm : 0.2f * mm);
  mm = xlv.z + xrv.z + a * wev.z; p += atv.z * (mm > 0.f ? mm : 0.2f * mm);
  mm = xlv.w + xrv.w + a * wev.w; p += atv.w * (mm > 0.f ? mm : 0.2f * mm);
#pragma unroll
  for (int off = 16; off > 0; off >>= 1) p += __shfl_xor(p, off, 32);
  if (lane == 0) { sbuf[e] = p; atomicMaxF(&smax[d], p); }
}

// Pass B: w = exp(s - smax[dst]); denom[dst] += w   (native f32 atomic add)
__global__ void edge_exp(const int* __restrict__ dst,
                         const float* __restrict__ sbuf,
                         const float* __restrict__ smax,
                         float* __restrict__ wbuf, float* __restrict__ denom) {
  const int e = blockIdx.x * blockDim.x + threadIdx.x;
  const int d = dst[e];
  const float w = __expf(sbuf[e] - smax[d]);
  wbuf[e] = w;
  atomicAdd(&denom[d], w);
}

// Pass C: agg[dst] += (w/denom[dst]) * xl[src]   (wave per edge, 4ch/lane)
__global__ __launch_bounds__(256) void edge_scatter(
    const int* __restrict__ src, const int* __restrict__ dst,
    const float* __restrict__ wbuf, const float* __restrict__ denom,
    const float* __restrict__ xl, float* __restrict__ agg) {
  const int e    = blockIdx.x * 8 + (threadIdx.x >> 5);
  const int lane = threadIdx.x & 31;
  const int s = src[e], d = dst[e];
  const float alpha = wbuf[e] / denom[d];
  const float4 xlv = *(const float4*)(xl + (size_t)s * HID + lane * 4);
  float* ap = agg + (size_t)d * HID + lane * 4;
  atomicAdd(ap + 0, alpha * xlv.x);
  atomicAdd(ap + 1, alpha * xlv.y);
  atomicAdd(ap + 2, alpha * xlv.z);
  atomicAdd(ap + 3, alpha * xlv.w);
}

// h = relu(agg + bias)
__global__ void node_finalize(const float* __restrict__ agg,
                              const float* __restrict__ bias,
                              float* __restrict__ h) {
  const int i = blockIdx.x * blockDim.x + threadIdx.x;
  h[i] = fmaxf(agg[i] + bias[i & (HID - 1)], 0.f);
}

// Output heads: mu[n,:] = tmu[n]@muW2+mub2 ; std = exp(log_std) ; value head.
__global__ void heads(const float* __restrict__ tmu, const float* __restrict__ tv,
                      const float* __restrict__ muW2, const float* __restrict__ mub2,
                      const float* __restrict__ vW2, const float* __restrict__ vb2,
                      const float* __restrict__ logstd, float* __restrict__ out) {
  const int n = blockIdx.x * blockDim.x + threadIdx.x;
  if (n >= NN) return;
  float accm[ACT];
#pragma unroll
  for (int a = 0; a < ACT; ++a) accm[a] = mub2[a];
  const float* tm = tmu + (size_t)n * HID;
  for (int c = 0; c < HID; ++c) {
    const float t = tm[c];
#pragma unroll
    for (int a = 0; a < ACT; ++a) accm[a] += t * muW2[c * ACT + a];
  }
  float* mu  = out;
  float* sd  = out + (size_t)NN * ACT;
  float* val = out + (size_t)2 * NN * ACT;
#pragma unroll
  for (int a = 0; a < ACT; ++a) mu[(size_t)n * ACT + a] = accm[a];
#pragma unroll
  for (int a = 0; a < ACT; ++a) sd[(size_t)n * ACT + a] = __expf(logstd[a]);
  const float* tr = tv + (size_t)n * HID;
  float acc = vb2[0];
  for (int c = 0; c < HID; ++c) acc += tr[c] * vW2[c];
  val[n] = acc;
}

extern "C" void kernel_launch(void* const* d_in, const int* in_sizes, int n_in,
                              void* d_out, int out_size, void* d_ws, size_t ws_size,
                              hipStream_t stream) {
  const float* x     = (const float*)d_in[0];
  const int*   ei    = (const int*)d_in[1];          // [2, NE] int32
  const int*   srcI  = ei;
  const int*   dstI  = ei + NE;
  const float* eattr = (const float*)d_in[2];
  // params in setup_inputs dict order
  const float *c1Wl = (const float*)d_in[3],  *c1bl = (const float*)d_in[4];
  const float *c1Wr = (const float*)d_in[5],  *c1br = (const float*)d_in[6];
  const float *c1We = (const float*)d_in[7],  *c1at = (const float*)d_in[8];
  const float *c1b  = (const float*)d_in[9];
  const float *c2Wl = (const float*)d_in[10], *c2bl = (const float*)d_in[11];
  const float *c2Wr = (const float*)d_in[12], *c2br = (const float*)d_in[13];
  const float *c2We = (const float*)d_in[14], *c2at = (const float*)d_in[15];
  const float *c2b  = (const float*)d_in[16];
  const float *muW1 = (const float*)d_in[17], *mub1 = (const float*)d_in[18];
  const float *muW2 = (const float*)d_in[19], *mub2 = (const float*)d_in[20];
  const float *vW1  = (const float*)d_in[21], *vb1  = (const float*)d_in[22];
  const float *vW2  = (const float*)d_in[23], *vb2  = (const float*)d_in[24];
  const float *lstd = (const float*)d_in[25];

  float* ws = (float*)d_ws;
  const size_t SZ = (size_t)NN * HID;               // 6.4M floats
  float* W0 = ws;            // xl  (both layers) -> later t_mu
  float* W1 = ws + SZ;       // xr  (both layers) -> later t_v
  float* W2 = ws + 2 * SZ;   // agg (both layers)
  float* W3 = ws + 3 * SZ;   // h1 then h2
  float* sbuf  = ws + 4 * SZ;
  float* wbuf  = sbuf + NE;
  float* smax  = wbuf + NE;
  float* denom = smax + NN;

  const int gT  = NN / 16;          // 3125 GEMM row tiles (exact)
  const int gN  = (NN * HID) / 256; // 25000 (exact)
  const int gEw = NE / 8;           // 80000 wave-per-edge blocks (exact)
  const int gEt = NE / 256;         // 2500  thread-per-edge blocks (exact)

  // ---- conv1 ----
  gemm_wmma_f32<OBS, false><<<gT, 256, 0, stream>>>(x, c1Wl, c1bl, W0);
  gemm_wmma_f32<OBS, false><<<gT, 256, 0, stream>>>(x, c1Wr, c1br, W1);
  init_seg<<<gN, 256, 0, stream>>>(smax, denom, W2);
  edge_logits<<<gEw, 256, 0, stream>>>(srcI, dstI, eattr, W0, W1, c1We, c1at, sbuf, smax);
  edge_exp<<<gEt, 256, 0, stream>>>(dstI, sbuf, smax, wbuf, denom);
  edge_scatter<<<gEw, 256, 0, stream>>>(srcI, dstI, wbuf, denom, W0, W2);
  node_finalize<<<gN, 256, 0, stream>>>(W2, c1b, W3);                 // h1

  // ---- conv2 ----
  gemm_wmma_f32<HID, false><<<gT, 256, 0, stream>>>(W3, c2Wl, c2bl, W0);
  gemm_wmma_f32<HID, false><<<gT, 256, 0, stream>>>(W3, c2Wr, c2br, W1);
  init_seg<<<gN, 256, 0, stream>>>(smax, denom, W2);
  edge_logits<<<gEw, 256, 0, stream>>>(srcI, dstI, eattr, W0, W1, c2We, c2at, sbuf, smax);
  edge_exp<<<gEt, 256, 0, stream>>>(dstI, sbuf, smax, wbuf, denom);
  edge_scatter<<<gEw, 256, 0, stream>>>(srcI, dstI, wbuf, denom, W0, W2);
  node_finalize<<<gN, 256, 0, stream>>>(W2, c2b, W3);                 // h2

  // ---- heads ----
  gemm_wmma_f32<HID, true><<<gT, 256, 0, stream>>>(W3, muW1, mub1, W0); // t_mu
  gemm_wmma_f32<HID, true><<<gT, 256, 0, stream>>>(W3, vW1, vb1, W1);   // t_v
  heads<<<(NN + 255) / 256, 256, 0, stream>>>(W0, W1, muW2, mub2, vW2, vb2, lstd,
                                              (float*)d_out);
}